// DotProductAttention_9337258902459
// MI455X (gfx1250) — compile-verified
//
#include <hip/hip_runtime.h>

// Problem constants from the reference
#define B_  2
#define H_  8
#define L_  512
#define D_  64
#define DM_ 512   // H_*D_

typedef float v2f __attribute__((ext_vector_type(2)));
typedef float v8f __attribute__((ext_vector_type(8)));

__device__ __forceinline__ float fast_tanh(float x) {
#if __has_builtin(__builtin_amdgcn_tanhf)
    return __builtin_amdgcn_tanhf(x);
#else
    float y;
    asm("v_tanh_f32 %0, %1" : "=v"(y) : "v"(x));
    return y;
#endif
}

// ---------------------------------------------------------------------------
// Kernel 1: f32 WMMA projections.
//   Y = X @ W with X:(B*L, DM), W:(DM, DM), written out in (B,H,L,D) layout.
//   One wave (32 threads) computes one 16x16 output tile via
//   v_wmma_f32_16x16x4_f32, looping K in steps of 4.
//   blockIdx.z selects (query,Wq)->Qout  vs  (key,Wk)->Kout.
//
//   f32 WMMA operand layout (ISA 7.12.2):
//     A 16x4 : lanes 0-15 = rows M, VGPR0/1 = K0/K1; lanes 16-31 = K2/K3
//     B 4x16 : VGPR0 = row K0 (lanes 0-15) / K2 (lanes 16-31); VGPR1 = K1/K3
//     C 16x16: VGPR i = row i (lanes 0-15) / row i+8 (lanes 16-31)
// ---------------------------------------------------------------------------
__global__ __launch_bounds__(32)
void proj_wmma_kernel(const float* __restrict__ Xq, const float* __restrict__ Xk,
                      const float* __restrict__ Wq, const float* __restrict__ Wk,
                      float* __restrict__ Qout, float* __restrict__ Kout) {
    const int lane = threadIdx.x;         // 0..31
    const int m    = lane & 15;           // row (A) / col (B,D) within tile
    const int kh   = lane >> 4;           // K-half select (0: K{0,1}, 1: K{2,3})
    const int tileM = blockIdx.x;         // 0..63  over B*L = 1024 rows
    const int tileN = blockIdx.y;         // 0..31  over DM = 512 cols

    const float* __restrict__ X   = blockIdx.z ? Xk   : Xq;
    const float* __restrict__ W   = blockIdx.z ? Wk   : Wq;
    float*       __restrict__ Out = blockIdx.z ? Kout : Qout;

    const int row = tileM * 16 + m;       // A-matrix row (0..1023)
    const int col = tileN * 16 + m;       // B/D column  (0..511)

    v8f acc = {};
    for (int k = 0; k < DM_; k += 4) {
        // A fragment: 2 contiguous f32 per lane -> one b64 load
        v2f a = *(const v2f*)(X + (size_t)row * DM_ + k + 2 * kh);
        // B fragment: rows (k + 2*kh) and (k + 2*kh + 1) of W at this column
        v2f b;
        b.x = W[(size_t)(k + 2 * kh + 0) * DM_ + col];
        b.y = W[(size_t)(k + 2 * kh + 1) * DM_ + col];
        acc = __builtin_amdgcn_wmma_f32_16x16x4_f32(
            /*neg_a=*/false, a, /*neg_b=*/false, b,
            /*c_mod=*/(short)0, acc, /*reuse_a=*/false, /*reuse_b=*/false);
    }

    // Scatter D into (B, H, L, D) layout:  flat row r = b*L + l, col = h*D + d
    const int h = col >> 6;               // /64
    const int d = col & (D_ - 1);
#pragma unroll
    for (int i = 0; i < 8; ++i) {
        const int r  = tileM * 16 + kh * 8 + i;   // 0..1023
        const int b_ = r >> 9;                    // /512
        const int l  = r & (L_ - 1);
        Out[(((size_t)b_ * H_ + h) * L_ + l) * D_ + d] = acc[i];
    }
}

// ---------------------------------------------------------------------------
// Kernel 2: additive-attention scores.
//   scores[bh,q,k] = sum_d wv[d] * tanh(Q[bh,q,d] + K[bh,k,d])
//   Block = 256 threads handles a 64(q) x 64(k) tile for one (b,h).
//   Q/K tiles staged in LDS (row pad 65 -> conflict-free column reads).
//   Each thread owns a 4x4 register tile: 16 tanh + 16 fma per d-step.
// ---------------------------------------------------------------------------
__global__ __launch_bounds__(256)
void score_kernel(const float* __restrict__ Qg, const float* __restrict__ Kg,
                  const float* __restrict__ wv, float* __restrict__ out) {
    __shared__ float Qs[64][65];
    __shared__ float Ks[64][65];
    __shared__ float wvs[64];

    const int tid = threadIdx.x;
    const int bh  = blockIdx.z;   // 0..15  (b*H + h)
    const int qt  = blockIdx.y;   // 0..7
    const int kt  = blockIdx.x;   // 0..7

    const float* __restrict__ Qtile = Qg + ((size_t)bh * L_ + qt * 64) * D_;
    const float* __restrict__ Ktile = Kg + ((size_t)bh * L_ + kt * 64) * D_;

    // Cooperative load: 64x64 floats each; 4 threads per row, float4 loads.
    {
        const int r  = tid >> 2;          // 0..63
        const int c0 = (tid & 3) * 16;    // 0,16,32,48
#pragma unroll
        for (int j = 0; j < 4; ++j) {
            const float4 q4 = *(const float4*)(Qtile + (size_t)r * D_ + c0 + j * 4);
            const float4 k4 = *(const float4*)(Ktile + (size_t)r * D_ + c0 + j * 4);
            Qs[r][c0 + j * 4 + 0] = q4.x; Qs[r][c0 + j * 4 + 1] = q4.y;
            Qs[r][c0 + j * 4 + 2] = q4.z; Qs[r][c0 + j * 4 + 3] = q4.w;
            Ks[r][c0 + j * 4 + 0] = k4.x; Ks[r][c0 + j * 4 + 1] = k4.y;
            Ks[r][c0 + j * 4 + 2] = k4.z; Ks[r][c0 + j * 4 + 3] = k4.w;
        }
        if (tid < 64) wvs[tid] = wv[tid];
    }
    __syncthreads();

    const int tx = tid & 15;   // k-group: columns tx*4 .. tx*4+3
    const int ty = tid >> 4;   // q-group: rows    ty*4 .. ty*4+3

    float acc[4][4] = {};
    for (int d = 0; d < D_; ++d) {
        const float w = wvs[d];
        float qv[4], kv[4];
#pragma unroll
        for (int i = 0; i < 4; ++i) qv[i] = Qs[ty * 4 + i][d];
#pragma unroll
        for (int j = 0; j < 4; ++j) kv[j] = Ks[tx * 4 + j][d];
#pragma unroll
        for (int i = 0; i < 4; ++i)
#pragma unroll
            for (int j = 0; j < 4; ++j)
                acc[i][j] = __builtin_fmaf(w, fast_tanh(qv[i] + kv[j]), acc[i][j]);
    }

    // Output (B,H,Lq,Lk): 4 consecutive k per thread -> aligned b128 stores.
#pragma unroll
    for (int i = 0; i < 4; ++i) {
        const int q = qt * 64 + ty * 4 + i;
        float4 o;
        o.x = acc[i][0]; o.y = acc[i][1]; o.z = acc[i][2]; o.w = acc[i][3];
        *(float4*)(out + ((size_t)bh * L_ + q) * L_ + kt * 64 + tx * 4) = o;
    }
}

extern "C" void kernel_launch(void* const* d_in, const int* in_sizes, int n_in,
                              void* d_out, int out_size, void* d_ws, size_t ws_size,
                              hipStream_t stream) {
    (void)in_sizes; (void)n_in; (void)out_size; (void)ws_size;
    const float* query = (const float*)d_in[0];
    const float* key   = (const float*)d_in[1];
    // d_in[2] = value : unused by the reference output
    const float* Wq    = (const float*)d_in[3];
    const float* Wk    = (const float*)d_in[4];
    // d_in[5] = Wv   : unused
    const float* wv    = (const float*)d_in[6];
    // d_in[7] = d_k  : unused (scores are un-scaled in the reference)

    float* out = (float*)d_out;
    float* Qws = (float*)d_ws;                              // (B,H,L,D) f32
    float* Kws = Qws + (size_t)B_ * H_ * L_ * D_;           // +2 MB

    // Projections via f32 WMMA: z=0 -> Q, z=1 -> K
    dim3 pgrid(64, 32, 2);
    proj_wmma_kernel<<<pgrid, 32, 0, stream>>>(query, key, Wq, Wk, Qws, Kws);

    // Additive scores
    dim3 sgrid(L_ / 64, L_ / 64, B_ * H_);
    score_kernel<<<sgrid, 256, 0, stream>>>(Qws, Kws, wv, out);
}